// E3NNFiducialCorrelator_84267258348034
// MI455X (gfx1250) — compile-verified
//
#include <hip/hip_runtime.h>
#include <math.h>

typedef float v2f __attribute__((ext_vector_type(2)));
typedef float v8f __attribute__((ext_vector_type(8)));

// ---------------- K (Clebsch-Gordan) tensor init, computed on device ----------------
struct Cplx { double re, im; };
__device__ inline Cplx cmul(Cplx a, Cplx b){ Cplx r; r.re=a.re*b.re-a.im*b.im; r.im=a.re*b.im+a.im*b.re; return r; }
__device__ double dfact(int n){ double r=1.0; for(int i=2;i<=n;++i) r*=(double)i; return r; }

__device__ double cgcoef(int j1,int m1,int j2,int m2,int j3,int m3){
  if (m1+m2 != m3) return 0.0;
  int dj = j1>j2 ? j1-j2 : j2-j1;
  if (j3 < dj || j3 > j1+j2) return 0.0;
  double pre = sqrt((2.0*j3+1.0)*dfact(j3+j1-j2)*dfact(j3-j1+j2)*dfact(j1+j2-j3)/dfact(j1+j2+j3+1));
  pre *= sqrt(dfact(j3+m3)*dfact(j3-m3)*dfact(j1-m1)*dfact(j1+m1)*dfact(j2-m2)*dfact(j2+m2));
  double s=0.0;
  for(int k=0;k<=j1+j2-j3;++k){
    int d1=j1+j2-j3-k, d2=j1-m1-k, d3=j2+m2-k, d4=j3-j2+m1+k, d5=j3-j1-m2+k;
    if (d1<0||d2<0||d3<0||d4<0||d5<0) continue;
    double den = dfact(k)*dfact(d1)*dfact(d2)*dfact(d3)*dfact(d4)*dfact(d5);
    s += ((k&1)? -1.0 : 1.0)/den;
  }
  return pre*s;
}

__device__ void buildQ(int l, Cplx Q[5][5]){
  for(int r=0;r<5;++r) for(int c=0;c<5;++c){ Q[r][c].re=0.0; Q[r][c].im=0.0; }
  Q[l][l].re = 1.0;
  const double s = 0.7071067811865476;
  for(int m=1;m<=l;++m){
    double sg = (m&1)? -1.0 : 1.0;
    Q[l+m][l+m].re = sg*s;
    Q[l+m][l-m].re = s;
    Q[l-m][l+m].im = -sg*s;
    Q[l-m][l-m].im = s;
  }
}

__device__ void computeK(int l1,int l2,int l3, float* out){
  Cplx Q1[5][5], Q2[5][5], Q3[5][5];
  buildQ(l1,Q1); buildQ(l2,Q2); buildQ(l3,Q3);
  int n1=2*l1+1, n2=2*l2+1, n3=2*l3+1;
  double re[125], im[125];
  double mr=0.0, mi=0.0;
  for(int c=0;c<n3;++c) for(int a=0;a<n1;++a) for(int b=0;b<n2;++b){
    Cplx acc; acc.re=0.0; acc.im=0.0;
    for(int C=0;C<n3;++C){
      Cplx q3 = Q3[c][C]; q3.im = -q3.im;
      for(int A=0;A<n1;++A){
        for(int B=0;B<n2;++B){
          double cgv = cgcoef(l1, A-l1, l2, B-l2, l3, C-l3);
          if (cgv==0.0) continue;
          Cplx t = cmul(q3, Q1[a][A]);
          t = cmul(t, Q2[b][B]);
          acc.re += t.re*cgv; acc.im += t.im*cgv;
        }
      }
    }
    int idx = (c*n1+a)*n2+b;
    re[idx]=acc.re; im[idx]=acc.im;
    if (fabs(acc.re)>mr) mr=fabs(acc.re);
    if (fabs(acc.im)>mi) mi=fabs(acc.im);
  }
  bool useIm = (mi > mr);
  int tot = n1*n2*n3;
  for(int i=0;i<tot;++i) out[i] = (float)(useIm ? im[i] : re[i]);
}

#define K121_OFF 0
#define K122_OFF 48
#define K221_OFF 128
#define K222_OFF 208
#define K_TOTAL  352

__global__ void init_k_kernel(float* __restrict__ K){
  int t = threadIdx.x;
  if      (t==0) computeK(1,2,1, K+K121_OFF);  // (3,3,5)
  else if (t==1) computeK(1,2,2, K+K122_OFF);  // (5,3,5)
  else if (t==2) computeK(2,2,1, K+K221_OFF);  // (3,5,5)
  else if (t==3) computeK(2,2,2, K+K222_OFF);  // (5,5,5)
}

// ---------------- KNN (top-9 incl self) + per-edge sh2 ----------------
__global__ void knn_sh_kernel(const float* __restrict__ pos, int* __restrict__ dst,
                              float* __restrict__ sh2, int N){
  __shared__ float tile[256*3];
  int i = blockIdx.x*256 + threadIdx.x;
  float px=0.f,py=0.f,pz=0.f;
  if (i<N){ px=pos[3*i]; py=pos[3*i+1]; pz=pos[3*i+2]; }
  float bd[9]; int bi[9];
  #pragma unroll
  for(int p=0;p<9;++p){ bd[p]=3.0e38f; bi[p]=0x7fffffff; }
  for(int j0=0;j0<N;j0+=256){
    __syncthreads();
    for(int t=threadIdx.x;t<768;t+=256)
      tile[t] = (j0 + t/3 < N) ? pos[(size_t)j0*3 + t] : 3.0e37f;
    __syncthreads();
    if (j0+256 < N) __builtin_prefetch(pos + (size_t)(j0+256)*3, 0, 1);
    if (i<N){
      for(int jj=0;jj<256;++jj){
        float dx=px-tile[3*jj], dy=py-tile[3*jj+1], dz=pz-tile[3*jj+2];
        float d = dx*dx+dy*dy+dz*dz;
        int j = j0+jj;
        if (d < bd[8] || (d==bd[8] && j<bi[8])){
          bd[8]=d; bi[8]=j;
          #pragma unroll
          for(int p=8;p>0;--p){
            bool sw = (bd[p]<bd[p-1]) || (bd[p]==bd[p-1] && bi[p]<bi[p-1]);
            if (sw){ float td=bd[p]; bd[p]=bd[p-1]; bd[p-1]=td;
                     int ti=bi[p]; bi[p]=bi[p-1]; bi[p-1]=ti; }
          }
        }
      }
    }
  }
  if (i>=N) return;
  const float s15 = 3.8729833462074170f, s5 = 2.2360679774997896f;
  #pragma unroll
  for(int k=0;k<8;++k){
    int j = bi[k+1];                      // drop self (slot 0)
    dst[i*8+k] = j;
    float ex = pos[3*j]-px, ey=pos[3*j+1]-py, ez=pos[3*j+2]-pz;
    float nrm = sqrtf(ex*ex+ey*ey+ez*ez) + 1e-12f;
    float x=ex/nrm, y=ey/nrm, z=ez/nrm;
    float* o = sh2 + (size_t)(i*8+k)*5;
    o[0]=s15*x*y; o[1]=s15*y*z; o[2]=0.5f*s5*(3.f*z*z-1.f);
    o[3]=s15*x*z; o[4]=0.5f*s15*(x*x-y*y);
  }
}

// ---------------- utility ----------------
__global__ void zero_kernel(float* __restrict__ p, long n){
  long t = (long)blockIdx.x*blockDim.x + threadIdx.x;
  long stride = (long)gridDim.x*blockDim.x;
  for(; t<n; t+=stride) p[t]=0.f;
}
__global__ void x0_init_kernel(float* __restrict__ x0, const float* __restrict__ Wemb, int n32){
  int t = blockIdx.x*blockDim.x + threadIdx.x;
  if (t<n32) x0[t] = Wemb[t & 31];
}

// ---------------- edge tensor-product + scatter (64 lanes / edge) ----------------
__global__ void edge_kernel(const float* __restrict__ x0, const float* __restrict__ x1,
                            const float* __restrict__ x2, const int* __restrict__ dst,
                            const float* __restrict__ sh2buf, const float* __restrict__ Kc,
                            const float* __restrict__ tpw,
                            float* __restrict__ a0, float* __restrict__ a1, float* __restrict__ a2,
                            int E){
  __shared__ float Ks[K_TOTAL];
  __shared__ float Wt[144];
  for(int t=threadIdx.x;t<K_TOTAL;t+=blockDim.x) Ks[t]=Kc[t];
  for(int t=threadIdx.x;t<144;t+=blockDim.x) Wt[t]=tpw[t];
  __syncthreads();
  int e = blockIdx.x*4 + (threadIdx.x>>6);
  int lane = threadIdx.x & 63;
  if (e>=E) return;
  int s = e>>3;
  int d = dst[e];
  float sh[5];
  #pragma unroll
  for(int j=0;j<5;++j) sh[j]=sh2buf[(size_t)e*5+j];

  if (lane<32){                         // x0 channels: paths p1 (K000) , p2 (K022=I)
    int u=lane;
    float xs = x0[(size_t)s*32+u];
    atomicAdd(&a0[(size_t)d*40+u], xs*Wt[u]);                 // p1 -> m0[0:32]
    float t2 = xs*Wt[32+u];
    #pragma unroll
    for(int o=0;o<5;++o) atomicAdd(&a2[((size_t)d*64+u)*5+o], t2*sh[o]);   // p2 -> m2[0:32]
  } else if (lane<48){                  // x1 channels: p3 (K101=I), p4 (K121), p5 (K122)
    int u=lane-32;
    float xi[3];
    #pragma unroll
    for(int i=0;i<3;++i) xi[i]=x1[((size_t)s*16+u)*3+i];
    float w3=Wt[64+u], w4=Wt[80+u], w5=Wt[96+u];
    #pragma unroll
    for(int i=0;i<3;++i) atomicAdd(&a1[((size_t)d*40+u)*3+i], xi[i]*w3);   // p3 -> m1[0:16]
    #pragma unroll
    for(int o=0;o<3;++o){
      float acc=0.f;
      #pragma unroll
      for(int i=0;i<3;++i){
        float xv=xi[i];
        #pragma unroll
        for(int j=0;j<5;++j) acc += xv*sh[j]*Ks[K121_OFF+(o*3+i)*5+j];
      }
      atomicAdd(&a1[((size_t)d*40+16+u)*3+o], acc*w4);                      // p4 -> m1[16:32]
    }
    #pragma unroll
    for(int o=0;o<5;++o){
      float acc=0.f;
      #pragma unroll
      for(int i=0;i<3;++i){
        float xv=xi[i];
        #pragma unroll
        for(int j=0;j<5;++j) acc += xv*sh[j]*Ks[K122_OFF+(o*3+i)*5+j];
      }
      atomicAdd(&a2[((size_t)d*64+32+u)*5+o], acc*w5);                      // p5 -> m2[32:48]
    }
  } else if (lane<56){                  // x2 channels: p6 (K202=I), p7 (K220=I/sqrt5), p8, p9
    int u=lane-48;
    float xi[5];
    #pragma unroll
    for(int i=0;i<5;++i) xi[i]=x2[((size_t)s*8+u)*5+i];
    float w6=Wt[112+u], w7=Wt[120+u], w8=Wt[128+u], w9=Wt[136+u];
    #pragma unroll
    for(int i=0;i<5;++i) atomicAdd(&a2[((size_t)d*64+48+u)*5+i], xi[i]*w6); // p6 -> m2[48:56]
    float p7=0.f;
    #pragma unroll
    for(int i=0;i<5;++i) p7 += xi[i]*sh[i];
    atomicAdd(&a0[(size_t)d*40+32+u], p7*0.44721359549995793f*w7);          // p7 -> m0[32:40]
    #pragma unroll
    for(int o=0;o<3;++o){
      float acc=0.f;
      #pragma unroll
      for(int i=0;i<5;++i){
        float xv=xi[i];
        #pragma unroll
        for(int j=0;j<5;++j) acc += xv*sh[j]*Ks[K221_OFF+(o*5+i)*5+j];
      }
      atomicAdd(&a1[((size_t)d*40+32+u)*3+o], acc*w8);                      // p8 -> m1[32:40]
    }
    #pragma unroll
    for(int o=0;o<5;++o){
      float acc=0.f;
      #pragma unroll
      for(int i=0;i<5;++i){
        float xv=xi[i];
        #pragma unroll
        for(int j=0;j<5;++j) acc += xv*sh[j]*Ks[K222_OFF+(o*5+i)*5+j];
      }
      atomicAdd(&a2[((size_t)d*64+56+u)*5+o], acc*w9);                      // p9 -> m2[56:64]
    }
  }
}

// ---------------- WMMA linear layers (V_WMMA_F32_16X16X4_F32, exact f32) ----------------
// y0 = a0 (N x 40) @ W0 (40 x 32) * (1/sqrt(40))
__global__ void wmma_lin0(const float* __restrict__ A, const float* __restrict__ W,
                          float* __restrict__ Y, int N){
  int w = blockIdx.x*(blockDim.x>>5) + (threadIdx.x>>5);
  int lane = threadIdx.x & 31;
  int mt = w >> 1, nt = w & 1;
  if (mt*16 >= N) return;
  int r16 = lane & 15, half = lane >> 4;
  int row = mt*16 + r16;
  int col = nt*16 + r16;
  v8f c = {0.f,0.f,0.f,0.f,0.f,0.f,0.f,0.f};
  for(int k=0;k<40;k+=4){
    int ka = k + 2*half;
    v2f a, b;
    a.x = A[(size_t)row*40 + ka];
    a.y = A[(size_t)row*40 + ka + 1];
    b.x = W[(size_t)ka*32 + col];
    b.y = W[(size_t)(ka+1)*32 + col];
    c = __builtin_amdgcn_wmma_f32_16x16x4_f32(false, a, false, b, (short)0, c, false, false);
  }
  const float sc = 0.15811388300841897f;
  #pragma unroll
  for(int r=0;r<8;++r){
    int orow = mt*16 + r + 8*half;
    Y[(size_t)orow*32 + col] = c[r]*sc;
  }
}

// y1[n,v,m] = sum_u a1[n,u,m] * W1[u,v] * (1/sqrt(40));  a1: (N,40,3), W1: (40,16)
__global__ void wmma_lin1(const float* __restrict__ A, const float* __restrict__ W,
                          float* __restrict__ Y, int N){
  int mt = blockIdx.x*(blockDim.x>>5) + (threadIdx.x>>5);
  int lane = threadIdx.x & 31;
  if (mt*16 >= N) return;
  int r16 = lane & 15, half = lane >> 4;
  int row = mt*16 + r16;
  int col = r16;
  const float sc = 0.15811388300841897f;
  for(int m=0;m<3;++m){
    v8f c = {0.f,0.f,0.f,0.f,0.f,0.f,0.f,0.f};
    for(int k=0;k<40;k+=4){
      int ka = k + 2*half;
      v2f a, b;
      a.x = A[((size_t)row*40 + ka)*3 + m];
      a.y = A[((size_t)row*40 + ka + 1)*3 + m];
      b.x = W[(size_t)ka*16 + col];
      b.y = W[(size_t)(ka+1)*16 + col];
      c = __builtin_amdgcn_wmma_f32_16x16x4_f32(false, a, false, b, (short)0, c, false, false);
    }
    #pragma unroll
    for(int r=0;r<8;++r){
      int orow = mt*16 + r + 8*half;
      Y[((size_t)orow*16 + col)*3 + m] = c[r]*sc;
    }
  }
}

// y2[n,v,m] = sum_u a2[n,u,m] * W2[u,v] * (1/8);  a2: (N,64,5), W2: (64,8)
__global__ void wmma_lin2(const float* __restrict__ A, const float* __restrict__ W,
                          float* __restrict__ Y, int N){
  int mt = blockIdx.x*(blockDim.x>>5) + (threadIdx.x>>5);
  int lane = threadIdx.x & 31;
  if (mt*16 >= N) return;
  int r16 = lane & 15, half = lane >> 4;
  int row = mt*16 + r16;
  int col = r16;
  for(int m=0;m<5;++m){
    v8f c = {0.f,0.f,0.f,0.f,0.f,0.f,0.f,0.f};
    for(int k=0;k<64;k+=4){
      int ka = k + 2*half;
      v2f a, b;
      a.x = A[((size_t)row*64 + ka)*5 + m];
      a.y = A[((size_t)row*64 + ka + 1)*5 + m];
      b.x = (col<8) ? W[(size_t)ka*8 + col]     : 0.f;
      b.y = (col<8) ? W[(size_t)(ka+1)*8 + col] : 0.f;
      c = __builtin_amdgcn_wmma_f32_16x16x4_f32(false, a, false, b, (short)0, c, false, false);
    }
    if (col<8){
      #pragma unroll
      for(int r=0;r<8;++r){
        int orow = mt*16 + r + 8*half;
        Y[((size_t)orow*8 + col)*5 + m] = c[r]*0.125f;
      }
    }
  }
}

// h1 = relu(x0 @ Wf1 * 1/sqrt(32));  (N,32)@(32,32)
__global__ void wmma_mlp1(const float* __restrict__ A, const float* __restrict__ W,
                          float* __restrict__ Y, int N){
  int w = blockIdx.x*(blockDim.x>>5) + (threadIdx.x>>5);
  int lane = threadIdx.x & 31;
  int mt = w >> 1, nt = w & 1;
  if (mt*16 >= N) return;
  int r16 = lane & 15, half = lane >> 4;
  int row = mt*16 + r16;
  int col = nt*16 + r16;
  v8f c = {0.f,0.f,0.f,0.f,0.f,0.f,0.f,0.f};
  for(int k=0;k<32;k+=4){
    int ka = k + 2*half;
    v2f a, b;
    a.x = A[(size_t)row*32 + ka];
    a.y = A[(size_t)row*32 + ka + 1];
    b.x = W[(size_t)ka*32 + col];
    b.y = W[(size_t)(ka+1)*32 + col];
    c = __builtin_amdgcn_wmma_f32_16x16x4_f32(false, a, false, b, (short)0, c, false, false);
  }
  const float sc = 0.17677669529663687f;
  #pragma unroll
  for(int r=0;r<8;++r){
    int orow = mt*16 + r + 8*half;
    Y[(size_t)orow*32 + col] = fmaxf(c[r]*sc, 0.f);
  }
}

// h2 = relu(h1 @ Wf2 + bf2);  (N,32)@(32,16)
__global__ void wmma_mlp2(const float* __restrict__ A, const float* __restrict__ W,
                          const float* __restrict__ bias, float* __restrict__ Y, int N){
  int mt = blockIdx.x*(blockDim.x>>5) + (threadIdx.x>>5);
  int lane = threadIdx.x & 31;
  if (mt*16 >= N) return;
  int r16 = lane & 15, half = lane >> 4;
  int row = mt*16 + r16;
  int col = r16;
  v8f c = {0.f,0.f,0.f,0.f,0.f,0.f,0.f,0.f};
  for(int k=0;k<32;k+=4){
    int ka = k + 2*half;
    v2f a, b;
    a.x = A[(size_t)row*32 + ka];
    a.y = A[(size_t)row*32 + ka + 1];
    b.x = W[(size_t)ka*16 + col];
    b.y = W[(size_t)(ka+1)*16 + col];
    c = __builtin_amdgcn_wmma_f32_16x16x4_f32(false, a, false, b, (short)0, c, false, false);
  }
  float bv = bias[col];
  #pragma unroll
  for(int r=0;r<8;++r){
    int orow = mt*16 + r + 8*half;
    Y[(size_t)orow*16 + col] = fmaxf(c[r] + bv, 0.f);
  }
}

// ---------------- batch-norm statistics ----------------
__global__ void stats0_kernel(const float* __restrict__ Y, float* __restrict__ stats, int N){
  __shared__ float sS[256], sQ[256];
  int c = threadIdx.x & 31;
  int ry = threadIdx.x >> 5;
  float s=0.f, q=0.f;
  for(int n = blockIdx.x*8 + ry; n < N; n += gridDim.x*8){
    float v = Y[(size_t)n*32+c]; s+=v; q+=v*v;
  }
  sS[threadIdx.x]=s; sQ[threadIdx.x]=q;
  __syncthreads();
  if (ry==0){
    for(int r=1;r<8;++r){ s+=sS[r*32+c]; q+=sQ[r*32+c]; }
    atomicAdd(&stats[c], s);
    atomicAdd(&stats[32+c], q);
  }
}
__global__ void stats1_kernel(const float* __restrict__ Y, float* __restrict__ stats, int N){
  __shared__ float sQ[256];
  int u = threadIdx.x & 15;
  int ry = threadIdx.x >> 4;
  float q=0.f;
  for(int n = blockIdx.x*16 + ry; n < N; n += gridDim.x*16){
    #pragma unroll
    for(int m=0;m<3;++m){ float v = Y[((size_t)n*16+u)*3+m]; q+=v*v; }
  }
  sQ[threadIdx.x]=q;
  __syncthreads();
  if (ry==0){
    for(int r=1;r<16;++r) q+=sQ[r*16+u];
    atomicAdd(&stats[64+u], q);
  }
}
__global__ void stats2_kernel(const float* __restrict__ Y, float* __restrict__ stats, int N){
  __shared__ float sQ[256];
  int u = threadIdx.x & 7;
  int ry = threadIdx.x >> 3;
  float q=0.f;
  for(int n = blockIdx.x*32 + ry; n < N; n += gridDim.x*32){
    #pragma unroll
    for(int m=0;m<5;++m){ float v = Y[((size_t)n*8+u)*5+m]; q+=v*v; }
  }
  sQ[threadIdx.x]=q;
  __syncthreads();
  if (ry==0){
    for(int r=1;r<32;++r) q+=sQ[r*8+u];
    atomicAdd(&stats[80+u], q);
  }
}

// ---------------- normalize + relu + residual ----------------
__global__ void apply_kernel(float* __restrict__ x0, float* __restrict__ x1, float* __restrict__ x2,
                             const float* __restrict__ y0, const float* __restrict__ y1,
                             const float* __restrict__ y2, const float* __restrict__ stats,
                             const float* __restrict__ bw0, const float* __restrict__ bb0,
                             const float* __restrict__ bw1, const float* __restrict__ bw2, int N){
  int n = blockIdx.x*blockDim.x + threadIdx.x;
  if (n>=N) return;
  float invN = 1.0f/(float)N;
  #pragma unroll
  for(int c=0;c<32;++c){
    float mean = stats[c]*invN;
    float var  = stats[32+c]*invN - mean*mean;     // mean((y-mu)^2)
    float v = (y0[(size_t)n*32+c]-mean)*rsqrtf(var+1e-5f)*bw0[c] + bb0[c];
    x0[(size_t)n*32+c] += fmaxf(v, 0.f);
  }
  #pragma unroll
  for(int u=0;u<16;++u){
    float g = rsqrtf(stats[64+u]*invN*(1.f/3.f) + 1e-5f)*bw1[u];
    #pragma unroll
    for(int m=0;m<3;++m){
      size_t idx=((size_t)n*16+u)*3+m;
      x1[idx] += fmaxf(y1[idx]*g, 0.f);
    }
  }
  #pragma unroll
  for(int u=0;u<8;++u){
    float g = rsqrtf(stats[80+u]*invN*0.2f + 1e-5f)*bw2[u];
    #pragma unroll
    for(int m=0;m<5;++m){
      size_t idx=((size_t)n*8+u)*5+m;
      x2[idx] += fmaxf(y2[idx]*g, 0.f);
    }
  }
}

// ---------------- head ----------------
__global__ void out_kernel(const float* __restrict__ h2, const float* __restrict__ Wf3,
                           const float* __restrict__ bf3, float* __restrict__ out, int N){
  int n = blockIdx.x*blockDim.x + threadIdx.x;
  if (n>=N) return;
  float o0=bf3[0], o1=bf3[1];
  #pragma unroll
  for(int k=0;k<16;++k){
    float h = h2[(size_t)n*16+k];
    o0 += h*Wf3[k*2];
    o1 += h*Wf3[k*2+1];
  }
  out[(size_t)n*2]   = o0;
  out[(size_t)n*2+1] = o1;
}

// ---------------- launch ----------------
extern "C" void kernel_launch(void* const* d_in, const int* in_sizes, int n_in,
                              void* d_out, int out_size, void* d_ws, size_t ws_size,
                              hipStream_t stream){
  const float* positions = (const float*)d_in[0];
  const float* W_emb  = (const float*)d_in[1];
  const float* tp_w   = (const float*)d_in[2];
  const float* lin_W0 = (const float*)d_in[3];
  const float* lin_W1 = (const float*)d_in[4];
  const float* lin_W2 = (const float*)d_in[5];
  const float* bn_w0  = (const float*)d_in[6];
  const float* bn_b0  = (const float*)d_in[7];
  const float* bn_w1  = (const float*)d_in[8];
  const float* bn_w2  = (const float*)d_in[9];
  const float* Wf1    = (const float*)d_in[10];
  const float* Wf2    = (const float*)d_in[11];
  const float* bf2    = (const float*)d_in[12];
  const float* Wf3    = (const float*)d_in[13];
  const float* bf3    = (const float*)d_in[14];
  float* out = (float*)d_out;
  (void)n_in; (void)out_size; (void)ws_size;

  int N = in_sizes[0]/3;      // 8192
  int E = N*8;

  float* ws = (float*)d_ws;
  size_t off = 0;
  float* K      = ws + off; off += K_TOTAL;
  int*   dstIdx = (int*)(ws + off); off += (size_t)E;
  float* sh2    = ws + off; off += (size_t)E*5;
  float* x0     = ws + off; off += (size_t)N*32;
  float* x1     = ws + off; off += (size_t)N*48;
  float* x2     = ws + off; off += (size_t)N*40;
  float* a0     = ws + off; off += (size_t)N*40;    // a0,a1,a2 contiguous
  float* a1     = ws + off; off += (size_t)N*120;
  float* a2     = ws + off; off += (size_t)N*320;
  float* y0     = ws + off; off += (size_t)N*32;
  float* y1     = ws + off; off += (size_t)N*48;
  float* y2     = ws + off; off += (size_t)N*40;
  float* stats  = ws + off; off += 128;
  float* h1     = ws + off; off += (size_t)N*32;
  float* h2     = ws + off; off += (size_t)N*16;

  init_k_kernel<<<1, 32, 0, stream>>>(K);
  knn_sh_kernel<<<(N+255)/256, 256, 0, stream>>>(positions, dstIdx, sh2, N);
  x0_init_kernel<<<(N*32+255)/256, 256, 0, stream>>>(x0, W_emb, N*32);
  zero_kernel<<<256, 256, 0, stream>>>(x1, (long)N*88);   // x1 + x2 contiguous

  int mt = N/16;
  for(int L=0; L<3; ++L){
    zero_kernel<<<512, 256, 0, stream>>>(a0, (long)N*480);        // a0+a1+a2
    zero_kernel<<<1, 128, 0, stream>>>(stats, 128);
    edge_kernel<<<(E+3)/4, 256, 0, stream>>>(x0,x1,x2,dstIdx,sh2,K,tp_w + (size_t)L*144,
                                             a0,a1,a2, E);
    wmma_lin0<<<(mt*2+7)/8, 256, 0, stream>>>(a0, lin_W0 + (size_t)L*1280, y0, N);
    wmma_lin1<<<(mt+7)/8,   256, 0, stream>>>(a1, lin_W1 + (size_t)L*640,  y1, N);
    wmma_lin2<<<(mt+7)/8,   256, 0, stream>>>(a2, lin_W2 + (size_t)L*512,  y2, N);
    stats0_kernel<<<64, 256, 0, stream>>>(y0, stats, N);
    stats1_kernel<<<64, 256, 0, stream>>>(y1, stats, N);
    stats2_kernel<<<64, 256, 0, stream>>>(y2, stats, N);
    apply_kernel<<<(N+255)/256, 256, 0, stream>>>(x0,x1,x2,y0,y1,y2,stats,
        bn_w0 + (size_t)L*32, bn_b0 + (size_t)L*32, bn_w1 + (size_t)L*16, bn_w2 + (size_t)L*8, N);
  }
  wmma_mlp1<<<(mt*2+7)/8, 256, 0, stream>>>(x0, Wf1, h1, N);
  wmma_mlp2<<<(mt+7)/8,   256, 0, stream>>>(h1, Wf2, bf2, h2, N);
  out_kernel<<<(N+255)/256, 256, 0, stream>>>(h2, Wf3, bf3, out, N);
}